// GNNx2_82222853914666
// MI455X (gfx1250) — compile-verified
//
#include <hip/hip_runtime.h>
#include <math.h>

#define NN 50000
#define NE 800000
#define D_EDGE 8
#define D_IN 128
#define D_HID 256
#define D_OUT 256

typedef __attribute__((ext_vector_type(2))) float v2f;
typedef __attribute__((ext_vector_type(8))) float v8f;

// ---------------- edge weight: softplus(ef @ We + be) ----------------
__global__ void edge_weight_kernel(const float* __restrict__ ef,
                                   const float* __restrict__ We,
                                   const float* __restrict__ be,
                                   float* __restrict__ w, int E) {
    int e = blockIdx.x * blockDim.x + threadIdx.x;
    if (e >= E) return;
    const float* f = ef + (size_t)e * D_EDGE;
    float acc = be[0];
#pragma unroll
    for (int k = 0; k < D_EDGE; ++k) acc += f[k] * We[k];
    w[e] = (acc > 20.f) ? acc : log1pf(expf(acc));
}

__global__ void fill_kernel(float* __restrict__ p, float v, int n) {
    int i = blockIdx.x * blockDim.x + threadIdx.x;
    if (i < n) p[i] = v;
}

__global__ void deg_scatter_kernel(const int* __restrict__ dst,
                                   const float* __restrict__ w,
                                   float* __restrict__ deg, int E) {
    int e = blockIdx.x * blockDim.x + threadIdx.x;
    if (e >= E) return;
    atomicAdd(deg + dst[e], w[e]);
}

__global__ void dinv_kernel(const float* __restrict__ deg,
                            float* __restrict__ dinv, int n) {
    int i = blockIdx.x * blockDim.x + threadIdx.x;
    if (i >= n) return;
    float d = deg[i];
    dinv[i] = (d > 0.f) ? rsqrtf(d) : 0.f;
}

__global__ void norm_kernel(const int* __restrict__ src, const int* __restrict__ dst,
                            const float* __restrict__ w, const float* __restrict__ dinv,
                            float* __restrict__ norm, int E) {
    int e = blockIdx.x * blockDim.x + threadIdx.x;
    if (e >= E) return;
    norm[e] = dinv[src[e]] * w[e] * dinv[dst[e]];
}

// broadcast bias row into every node row (agg buffer init; GCN adds bias post-sum)
__global__ void bias_fill_kernel(float* __restrict__ out, const float* __restrict__ bias,
                                 int total4, int d4) {
    int i = blockIdx.x * blockDim.x + threadIdx.x;
    if (i >= total4) return;
    ((float4*)out)[i] = ((const float4*)bias)[i % d4];
}

__global__ void tanh_kernel(float* __restrict__ x, int n4) {
    int i = blockIdx.x * blockDim.x + threadIdx.x;
    if (i >= n4) return;
    float4 v = ((float4*)x)[i];
    v.x = tanhf(v.x); v.y = tanhf(v.y); v.z = tanhf(v.z); v.w = tanhf(v.w);
    ((float4*)x)[i] = v;
}

// ------------- message passing: out[dst] += norm * h[src] -------------
// one block per segment; segments [0,E) are edges, [E,E+NN) are self loops.
__global__ void aggregate_kernel(const float* __restrict__ h,
                                 const int* __restrict__ src,
                                 const int* __restrict__ dst,
                                 const float* __restrict__ norm,
                                 const float* __restrict__ dinv,
                                 float* __restrict__ out,
                                 int E, int d) {
    int seg = blockIdx.x;
    int s, t; float w;
    if (seg < E) { s = src[seg]; t = dst[seg]; w = norm[seg]; }
    else { int i = seg - E; s = i; t = i; float di = dinv[i]; w = di * di; }
    int c = threadIdx.x * 4;               // blockDim.x == d/4
    const float4 hv = *(const float4*)(h + (size_t)s * d + c);
    float* o = out + (size_t)t * d + c;
    atomicAdd(o + 0, w * hv.x);
    atomicAdd(o + 1, w * hv.y);
    atomicAdd(o + 2, w * hv.z);
    atomicAdd(o + 3, w * hv.w);
}

// ------------- f32 GEMM via V_WMMA_F32_16X16X4_F32 -------------
// C[M,N] = A[M,K] @ B[K,N] (+ bias[N]) (+ addend[M,N])
// 128 threads = 4 waves; wave -> one 16-row tile; blockIdx.y -> 64-col strip.
__global__ __launch_bounds__(128)
void gemm_f32_wmma(const float* __restrict__ A, const float* __restrict__ B,
                   float* __restrict__ C, int M, int K, int N,
                   const float* __restrict__ bias, const float* __restrict__ addend) {
    const int lane = threadIdx.x & 31;
    const int wave = threadIdx.x >> 5;
    const int rowTile = blockIdx.x * 4 + wave;
    if (rowTile * 16 >= M) return;          // wave-uniform exit
    const int colBase = blockIdx.y * 64;
    const int half  = lane >> 4;            // 0: lanes 0-15, 1: lanes 16-31
    const int l15   = lane & 15;
    const int koff  = half * 2;             // A/B K sub-offset per ISA 16x4 layout
    const int mrow  = rowTile * 16 + l15;
    const int ncol  = colBase + l15;

    v8f c0 = {}, c1 = {}, c2 = {}, c3 = {};
    const float* Arow = A + (size_t)mrow * K;

    for (int k0 = 0; k0 < K; k0 += 4) {
        const float2 av = *(const float2*)(Arow + k0 + koff);   // K=koff, koff+1
        v2f a; a.x = av.x; a.y = av.y;
        const float* Bp = B + (size_t)(k0 + koff) * N + ncol;
        v2f b0, b1, b2, b3;
        b0.x = Bp[0];  b0.y = Bp[N];
        b1.x = Bp[16]; b1.y = Bp[N + 16];
        b2.x = Bp[32]; b2.y = Bp[N + 32];
        b3.x = Bp[48]; b3.y = Bp[N + 48];
        c0 = __builtin_amdgcn_wmma_f32_16x16x4_f32(false, a, false, b0, (short)0, c0, false, false);
        c1 = __builtin_amdgcn_wmma_f32_16x16x4_f32(false, a, false, b1, (short)0, c1, false, false);
        c2 = __builtin_amdgcn_wmma_f32_16x16x4_f32(false, a, false, b2, (short)0, c2, false, false);
        c3 = __builtin_amdgcn_wmma_f32_16x16x4_f32(false, a, false, b3, (short)0, c3, false, false);
    }

    // D layout: VGPR r, lanes 0-15 -> M=r, lanes 16-31 -> M=8+r; N = lane%16
    const int rbase = rowTile * 16 + (half ? 8 : 0);
    const int col = colBase + l15;
#pragma unroll
    for (int r = 0; r < 8; ++r) {
        const size_t o = (size_t)(rbase + r) * N + col;
        float v0 = c0[r], v1 = c1[r], v2 = c2[r], v3 = c3[r];
        if (bias)   { v0 += bias[col];     v1 += bias[col + 16];
                      v2 += bias[col + 32]; v3 += bias[col + 48]; }
        if (addend) { v0 += addend[o];      v1 += addend[o + 16];
                      v2 += addend[o + 32]; v3 += addend[o + 48]; }
        C[o] = v0; C[o + 16] = v1; C[o + 32] = v2; C[o + 48] = v3;
    }
}

extern "C" void kernel_launch(void* const* d_in, const int* in_sizes, int n_in,
                              void* d_out, int out_size, void* d_ws, size_t ws_size,
                              hipStream_t stream) {
    (void)in_sizes; (void)n_in; (void)out_size; (void)ws_size;
    const int*   edge_index = (const int*)d_in[0];     // [2, E]
    const float* edge_feats = (const float*)d_in[1];   // [E, 8]
    const float* X          = (const float*)d_in[2];   // [N, 128]
    const float* We         = (const float*)d_in[3];
    const float* be         = (const float*)d_in[4];
    const float* W1         = (const float*)d_in[5];
    const float* b1         = (const float*)d_in[6];
    const float* W2         = (const float*)d_in[7];
    const float* b2         = (const float*)d_in[8];
    const float* Ws         = (const float*)d_in[9];
    const float* bs         = (const float*)d_in[10];
    float* out = (float*)d_out;

    const int* src = edge_index;
    const int* dst = edge_index + NE;

    // workspace layout (floats)
    float* ws = (float*)d_ws;
    const size_t NODEF = (size_t)NN * D_HID;           // 12.8M floats
    float* bufA   = ws;                                 // [N, 256] gemm out
    float* bufB   = bufA + NODEF;                       // [N, 256] agg buffer
    float* edge_w = bufB + NODEF;                       // [E]
    float* norm   = edge_w + NE;                        // [E]
    float* deg    = norm + NE;                          // [N]
    float* dinv   = deg + NN;                           // [N]

    const int TB = 256;
    const dim3 gGemm((NN / 16 + 3) / 4, D_HID / 64);    // 782 x 4

    // --- normalization pipeline ---
    edge_weight_kernel<<<(NE + TB - 1) / TB, TB, 0, stream>>>(edge_feats, We, be, edge_w, NE);
    fill_kernel<<<(NN + TB - 1) / TB, TB, 0, stream>>>(deg, 1.0f, NN);   // self-loop weight
    deg_scatter_kernel<<<(NE + TB - 1) / TB, TB, 0, stream>>>(dst, edge_w, deg, NE);
    dinv_kernel<<<(NN + TB - 1) / TB, TB, 0, stream>>>(deg, dinv, NN);
    norm_kernel<<<(NE + TB - 1) / TB, TB, 0, stream>>>(src, dst, edge_w, dinv, norm, NE);

    // --- layer 1: h1 = X @ W1 ; agg1 = segsum + b1 ; t = tanh(agg1) ---
    gemm_f32_wmma<<<gGemm, 128, 0, stream>>>(X, W1, bufA, NN, D_IN, D_HID, nullptr, nullptr);
    bias_fill_kernel<<<(int)((NODEF / 4 + TB - 1) / TB), TB, 0, stream>>>(bufB, b1, (int)(NODEF / 4), D_HID / 4);
    aggregate_kernel<<<NE + NN, D_HID / 4, 0, stream>>>(bufA, src, dst, norm, dinv, bufB, NE, D_HID);
    tanh_kernel<<<(int)((NODEF / 4 + TB - 1) / TB), TB, 0, stream>>>(bufB, (int)(NODEF / 4));

    // --- layer 2: h2 = t @ W2 ; agg2 = segsum + b2 ---
    gemm_f32_wmma<<<gGemm, 128, 0, stream>>>(bufB, W2, bufA, NN, D_HID, D_OUT, nullptr, nullptr);
    bias_fill_kernel<<<(int)((NODEF / 4 + TB - 1) / TB), TB, 0, stream>>>(bufB, b2, (int)(NODEF / 4), D_OUT / 4);
    aggregate_kernel<<<NE + NN, D_OUT / 4, 0, stream>>>(bufA, src, dst, norm, dinv, bufB, NE, D_OUT);

    // --- fused residual: out = X @ Ws + bs + agg2 ---
    gemm_f32_wmma<<<gGemm, 128, 0, stream>>>(X, Ws, out, NN, D_IN, D_OUT, bs, bufB);
}